// CenterLoss_5239860101311
// MI455X (gfx1250) — compile-verified
//
#include <hip/hip_runtime.h>

typedef __attribute__((ext_vector_type(2))) float v2f;
typedef __attribute__((ext_vector_type(8))) float v8f;

#define BATCH     1024
#define NUM_CLASS 100000
#define FEAT      128

// Seed the closed-form constant: (C-1) entries per row clip to 1e-12, and the
// 1/B and B rows cancel -> (NUM_CLASS-1) * 1e-12 added once.
__global__ void CenterLoss_init(float* __restrict__ out) {
  if (threadIdx.x == 0) out[0] = (float)((double)(NUM_CLASS - 1) * 1e-12);
}

// One wave (32 lanes) per 16-row tile. S = X_tile (16x128) @ G_tile^T (128x16)
// accumulated over K in steps of 4 via V_WMMA_F32_16X16X4_F32; only diag(S)
// is consumed. Row norms computed from the same float2 loads.
__global__ __launch_bounds__(32)
void CenterLoss_wmma(const float* __restrict__ x,
                     const int*   __restrict__ labels,
                     const float* __restrict__ centers,
                     float* __restrict__ out) {
  const int lane    = threadIdx.x;   // 0..31, wave32
  const int idx     = lane & 15;     // row / col within tile
  const int half    = lane >> 4;     // K sub-lane group
  const int rowbase = blockIdx.x * 16;

  const int lab = labels[rowbase + idx];
  const float* __restrict__ xrow = x       + (size_t)(rowbase + idx) * FEAT;
  const float* __restrict__ crow = centers + (size_t)lab * FEAT;

  v8f acc = {};            // 16x16 f32 accumulator: S[m][n] = x_m . c_{l_n}
  float sx = 0.0f;         // partial ||x_row||^2 (this lane's K subset)
  float sc = 0.0f;         // partial ||c_row||^2

  // A 16x4 f32 layout: lanes 0-15 hold K = kb,kb+1 ; lanes 16-31 hold K = kb+2,kb+3
  // (VGPR0 = K even-of-pair, VGPR1 = K odd-of-pair). B (4x16) mirrors this.
#pragma unroll
  for (int kk = 0; kk < FEAT / 4; ++kk) {
    const int kb = kk * 4 + 2 * half;
    v2f a, b;
    a.x = xrow[kb];  a.y = xrow[kb + 1];
    b.x = crow[kb];  b.y = crow[kb + 1];
    sx += a.x * a.x + a.y * a.y;
    sc += b.x * b.x + b.y * b.y;
    // v_wmma_f32_16x16x4_f32: (neg_a, A, neg_b, B, c_mod, C, reuse_a, reuse_b)
    acc = __builtin_amdgcn_wmma_f32_16x16x4_f32(false, a, false, b,
                                                (short)0, acc, false, false);
  }

  // Lanes L and L^16 covered complementary K subsets of row (L&15):
  // combine to get full norms (valid on lanes 0..15).
  sx += __shfl_xor(sx, 16, 32);
  sc += __shfl_xor(sc, 16, 32);

  // Diagonal of S from the 16x16 f32 C/D layout:
  //   i in 0..7  : VGPR i,   lane i        (M=r, N=lane)
  //   i in 8..15 : VGPR i-8, lane i+16     (M=r+8, N=lane-16)
  const int sel = (lane < 8) ? lane : ((lane >= 24) ? (lane - 24) : -1);
  float mydiag = 0.0f;
#pragma unroll
  for (int r = 0; r < 8; ++r) {
    const float v = acc[r];
    mydiag = (sel == r) ? v : mydiag;
  }
  // Bring diag for rows 8..15 (held in lanes 24..31) down to lanes 8..15.
  const float diag = mydiag + __shfl_xor(mydiag, 16, 32);

  float contrib = 0.0f;
  if (lane < 16) {
    float d = sx + sc - 2.0f * diag;       // expansion form, faithful to ref
    d = fminf(fmaxf(d, 1e-12f), 1e12f);    // clamp masked entry
    contrib = d;
  }

  // Wave-wide sum (lanes >=16 contribute 0), then one atomic per tile.
#pragma unroll
  for (int m = 16; m >= 1; m >>= 1) contrib += __shfl_xor(contrib, m, 32);
  if (lane == 0) atomicAdd(out, contrib * (1.0f / (float)BATCH));
}

extern "C" void kernel_launch(void* const* d_in, const int* in_sizes, int n_in,
                              void* d_out, int out_size, void* d_ws, size_t ws_size,
                              hipStream_t stream) {
  const float* x       = (const float*)d_in[0];
  const int*   labels  = (const int*)d_in[1];
  const float* centers = (const float*)d_in[2];
  float* out = (float*)d_out;

  CenterLoss_init<<<1, 32, 0, stream>>>(out);
  CenterLoss_wmma<<<BATCH / 16, 32, 0, stream>>>(x, labels, centers, out);
}